// IndexDGCRN_79937931313173
// MI455X (gfx1250) — compile-verified
//
#include <hip/hip_runtime.h>
#include <math.h>

// ---------------------------------------------------------------------------
// IndexDGCRN on gfx1250: all GEMMs via V_WMMA_F32_16X16X4_F32 (fp32 matrix path)
// v2: 32x32 C-region per wave (4 accumulators), software-pipelined global->LDS
// ---------------------------------------------------------------------------

typedef __attribute__((ext_vector_type(2))) float v2f;
typedef __attribute__((ext_vector_type(8))) float v8f;

#define OP_NONE 0
#define OP_CHEB 1   // out = 2*acc - I   (Chebyshev T2 = 2*A*A - I)
#define OP_SIG  2   // out = sigmoid(acc + bias[col])
#define OP_TANH 3   // out = tanh(acc + bias[col])

#define BM 128
#define BN 64
#define BK 16

// Model constants (from reference setup_inputs)
#define BB   32
#define TT   12
#define NN   512
#define HH   64
#define EE   10
#define ENC_C 65    // D + H
#define ENC_LD 195  // K * ENC_C
#define DEC_C 66    // out_d + ycov + H
#define DEC_LD 198  // K * DEC_C

// ---------------------------------------------------------------------------
// Tiled WMMA FP32 GEMM:  C[bz] = epilogue( A[bz] @ B[bz] )
// 256 threads = 8 wave32. Block tile 128x64; each wave owns a 32x32 C region
// (2x2 of 16x16 WMMA tiles -> 2 A-frags x 2 B-frags feed 4 WMMAs per k-step).
// Global->LDS copies are software-pipelined through registers so HBM/L2
// latency overlaps the WMMA stream. Ragged M/N/K handled by zero-padded LDS
// tiles + guarded stores.
// ---------------------------------------------------------------------------
__global__ __launch_bounds__(256) void wmma_gemm_f32(
    const float* __restrict__ Ag, const float* __restrict__ Bg,
    float* __restrict__ Cg, const float* __restrict__ bias,
    int M, int Nd, int Kd, int lda, int ldb, int ldc,
    long long aBatch, long long bBatch, long long cBatch, int op)
{
  __shared__ float As[BM][BK + 1];
  __shared__ float Bs[BK][BN + 1];

  const int tid  = threadIdx.x;
  const int lane = tid & 31;
  const int wave = tid >> 5;
  const int half = lane >> 4;   // 0: lanes 0-15, 1: lanes 16-31
  const int lr   = lane & 15;

  const int wrow = wave & 3;    // 4 M-groups of 32 rows
  const int wcol = wave >> 2;   // 2 N-groups of 32 cols

  const int blockM = blockIdx.y * BM;
  const int blockN = blockIdx.x * BN;
  const int bz = blockIdx.z;

  const float* Ab = Ag + aBatch * bz;
  const float* Bb = Bg + bBatch * bz;
  float*       Cb = Cg + cBatch * bz;

  v8f acc00 = {};
  v8f acc01 = {};
  v8f acc10 = {};
  v8f acc11 = {};

  float aR[8];   // staging regs: A tile 128x16 = 2048 elems / 256 thr
  float bR[4];   // staging regs: B tile  16x64 = 1024 elems / 256 thr

  auto loadA = [&](int k0) {
#pragma unroll
    for (int i = 0; i < 8; ++i) {
      int idx = tid + i * 256;
      int m = idx >> 4, kk = idx & 15;
      int gm = blockM + m, gk = k0 + kk;
      aR[i] = (gm < M && gk < Kd) ? Ab[(long long)gm * lda + gk] : 0.0f;
    }
  };
  auto loadB = [&](int k0) {
#pragma unroll
    for (int i = 0; i < 4; ++i) {
      int idx = tid + i * 256;
      int kk = idx >> 6, n = idx & 63;
      int gk = k0 + kk, gn = blockN + n;
      bR[i] = (gk < Kd && gn < Nd) ? Bb[(long long)gk * ldb + gn] : 0.0f;
    }
  };

  // prologue: fetch first k-tile into registers
  loadA(0);
  loadB(0);

  for (int k0 = 0; k0 < Kd; k0 += BK) {
    // commit staged tile to LDS
#pragma unroll
    for (int i = 0; i < 8; ++i) {
      int idx = tid + i * 256;
      As[idx >> 4][idx & 15] = aR[i];
    }
#pragma unroll
    for (int i = 0; i < 4; ++i) {
      int idx = tid + i * 256;
      Bs[idx >> 6][idx & 63] = bR[i];
    }
    __syncthreads();

    // issue next tile's global loads early (overlap with WMMA stream)
    if (k0 + BK < Kd) {
      loadA(k0 + BK);
      loadB(k0 + BK);
    }

#pragma unroll
    for (int kk = 0; kk < BK; kk += 4) {
      // A frag 16x4: lane = M row (lr), halves hold K pairs {0,1}/{2,3}
      v2f a0, a1, b0, b1;
      a0.x = As[wrow * 32 + lr][kk + half * 2 + 0];
      a0.y = As[wrow * 32 + lr][kk + half * 2 + 1];
      a1.x = As[wrow * 32 + 16 + lr][kk + half * 2 + 0];
      a1.y = As[wrow * 32 + 16 + lr][kk + half * 2 + 1];
      // B frag 4x16: lane = N col (lr), halves hold K pairs
      b0.x = Bs[kk + half * 2 + 0][wcol * 32 + lr];
      b0.y = Bs[kk + half * 2 + 1][wcol * 32 + lr];
      b1.x = Bs[kk + half * 2 + 0][wcol * 32 + 16 + lr];
      b1.y = Bs[kk + half * 2 + 1][wcol * 32 + 16 + lr];
      acc00 = __builtin_amdgcn_wmma_f32_16x16x4_f32(false, a0, false, b0,
                                                    (short)0, acc00, false, false);
      acc01 = __builtin_amdgcn_wmma_f32_16x16x4_f32(false, a0, false, b1,
                                                    (short)0, acc01, false, false);
      acc10 = __builtin_amdgcn_wmma_f32_16x16x4_f32(false, a1, false, b0,
                                                    (short)0, acc10, false, false);
      acc11 = __builtin_amdgcn_wmma_f32_16x16x4_f32(false, a1, false, b1,
                                                    (short)0, acc11, false, false);
    }
    __syncthreads();
  }

  // Epilogue. C/D layout: VGPR v, lanes 0-15 -> M=v, lanes 16-31 -> M=v+8.
#pragma unroll
  for (int mi = 0; mi < 2; ++mi) {
#pragma unroll
    for (int nj = 0; nj < 2; ++nj) {
      v8f acc = (mi == 0) ? (nj == 0 ? acc00 : acc01)
                          : (nj == 0 ? acc10 : acc11);
      int gcol = blockN + wcol * 32 + nj * 16 + lr;
      if (gcol >= Nd) continue;
      float bv = (bias && op != OP_CHEB) ? bias[gcol] : 0.0f;
#pragma unroll
      for (int v = 0; v < 8; ++v) {
        int grow = blockM + wrow * 32 + mi * 16 + v + 8 * half;
        if (grow >= M) continue;
        float val = acc[v];
        if (op == OP_CHEB) {
          val = 2.0f * val - (grow == gcol ? 1.0f : 0.0f);
        } else {
          val += bv;
          if (op == OP_SIG)       val = 1.0f / (1.0f + __expf(-val));
          else if (op == OP_TANH) val = tanhf(val);
        }
        Cb[(long long)grow * ldc + gcol] = val;
      }
    }
  }
}

// ---------------------------------------------------------------------------
// Row softmax(relu(emb @ emb^T)) for one (batch,row): N=512, 256 threads.
// Used for both static graph (batch=1) and dynamic graphs (batch=32).
// ---------------------------------------------------------------------------
__global__ __launch_bounds__(256) void adj_softmax_kernel(
    const float* __restrict__ emb, float* __restrict__ Aout,
    long long embBatch, long long aBatch)
{
  const int row = blockIdx.x;
  const float* eb = emb + embBatch * blockIdx.y;
  float* Ab = Aout + aBatch * blockIdx.y;

  __shared__ float erow[EE];
  __shared__ float red[256];
  const int tid = threadIdx.x;
  if (tid < EE) erow[tid] = eb[(long long)row * EE + tid];
  __syncthreads();

  const int c0 = tid, c1 = tid + 256;
  float d0 = 0.0f, d1 = 0.0f;
#pragma unroll
  for (int e = 0; e < EE; ++e) {
    float er = erow[e];
    d0 += er * eb[(long long)c0 * EE + e];
    d1 += er * eb[(long long)c1 * EE + e];
  }
  float s0 = fmaxf(d0, 0.0f), s1 = fmaxf(d1, 0.0f);

  red[tid] = fmaxf(s0, s1);
  __syncthreads();
  for (int off = 128; off > 0; off >>= 1) {
    if (tid < off) red[tid] = fmaxf(red[tid], red[tid + off]);
    __syncthreads();
  }
  float m = red[0];
  __syncthreads();

  float e0 = __expf(s0 - m), e1 = __expf(s1 - m);
  red[tid] = e0 + e1;
  __syncthreads();
  for (int off = 128; off > 0; off >>= 1) {
    if (tid < off) red[tid] += red[tid + off];
    __syncthreads();
  }
  float inv = 1.0f / red[0];
  Ab[(long long)row * NN + c0] = e0 * inv;
  Ab[(long long)row * NN + c1] = e1 * inv;
}

// ---------------------------------------------------------------------------
// Encoder concat: xg[b,n,0:65] = [x[b,t,n,0], (cand? z*h : h)]
// Written directly into the k=0 slice of xg (identity Chebyshev term).
// ---------------------------------------------------------------------------
__global__ void enc_concat_kernel(
    const float* __restrict__ x, const float* __restrict__ h,
    const float* __restrict__ zr, float* __restrict__ xg,
    int t, int useCand)
{
  int idx = blockIdx.x * blockDim.x + threadIdx.x;
  const int total = BB * NN * ENC_C;
  if (idx >= total) return;
  int c = idx % ENC_C;
  int bn = idx / ENC_C;
  int b = bn / NN, n = bn % NN;
  float v;
  if (c == 0) {
    v = x[(((long long)b * TT + t) * NN + n)];
  } else {
    float hv = h[(long long)bn * HH + (c - 1)];
    if (useCand) hv *= zr[(long long)bn * (2 * HH) + (c - 1)];  // z gate
    v = hv;
  }
  xg[(long long)bn * ENC_LD + c] = v;
}

// ---------------------------------------------------------------------------
// Decoder concat: xg[b,n,0:66] = [go[b,n], ycov[b,t,n,0], (cand? z*h : h)]
// ---------------------------------------------------------------------------
__global__ void dec_concat_kernel(
    const float* __restrict__ go, const float* __restrict__ ycov,
    const float* __restrict__ h, const float* __restrict__ zr,
    float* __restrict__ xg, int t, int useCand)
{
  int idx = blockIdx.x * blockDim.x + threadIdx.x;
  const int total = BB * NN * DEC_C;
  if (idx >= total) return;
  int c = idx % DEC_C;
  int bn = idx / DEC_C;
  int b = bn / NN, n = bn % NN;
  float v;
  if (c == 0) {
    v = go[bn];
  } else if (c == 1) {
    v = ycov[(((long long)b * TT + t) * NN + n)];
  } else {
    float hv = h[(long long)bn * HH + (c - 2)];
    if (useCand) hv *= zr[(long long)bn * (2 * HH) + (c - 2)];
    v = hv;
  }
  xg[(long long)bn * DEC_LD + c] = v;
}

// h = r*h + (1-r)*hc ;  r = zr[..., H:2H]
__global__ void gru_combine_kernel(
    float* __restrict__ h, const float* __restrict__ zr,
    const float* __restrict__ hc)
{
  int idx = blockIdx.x * blockDim.x + threadIdx.x;
  const int total = BB * NN * HH;
  if (idx >= total) return;
  int j = idx % HH;
  int bn = idx / HH;
  float r = zr[(long long)bn * (2 * HH) + HH + j];
  h[idx] = r * h[idx] + (1.0f - r) * hc[idx];
}

// E_dyn[b,n,e] = node_emb[n,e] + hyper_b[e] + sum_k h[b,n,k]*hyper_W[k,e]
__global__ void edyn_kernel(
    const float* __restrict__ h, const float* __restrict__ hW,
    const float* __restrict__ hb, const float* __restrict__ ne,
    float* __restrict__ Ed)
{
  int idx = blockIdx.x * blockDim.x + threadIdx.x;
  const int total = BB * NN * EE;
  if (idx >= total) return;
  int e = idx % EE;
  int bn = idx / EE;
  int n = bn % NN;
  float acc = hb[e] + ne[(long long)n * EE + e];
  const float* hp = h + (long long)bn * HH;
#pragma unroll
  for (int k = 0; k < HH; ++k) acc += hp[k] * hW[k * EE + e];
  Ed[idx] = acc;
}

// go[b,n] = h[b,n,:] . proj_W + proj_b ; also out[b,t,n,0] = go
__global__ void proj_kernel(
    const float* __restrict__ h, const float* __restrict__ pW,
    const float* __restrict__ pb, float* __restrict__ go,
    float* __restrict__ out, int t)
{
  int idx = blockIdx.x * blockDim.x + threadIdx.x;
  if (idx >= BB * NN) return;
  int b = idx / NN, n = idx % NN;
  const float* hp = h + (long long)idx * HH;
  float acc = pb[0];
#pragma unroll
  for (int k = 0; k < HH; ++k) acc += hp[k] * pW[k];
  go[idx] = acc;
  out[(((long long)b * TT + t) * NN + n)] = acc;
}

// ---------------------------------------------------------------------------
// Host orchestration
// ---------------------------------------------------------------------------
static inline void launch_gemm(hipStream_t s,
                               const float* A, int lda, long long aB,
                               const float* B, int ldb, long long bB,
                               float* C, int ldc, long long cB,
                               int M, int Nd, int Kd, int batch, int op,
                               const float* bias)
{
  dim3 grid((Nd + BN - 1) / BN, (M + BM - 1) / BM, batch);
  wmma_gemm_f32<<<grid, 256, 0, s>>>(A, B, C, bias, M, Nd, Kd,
                                     lda, ldb, ldc, aB, bB, cB, op);
}

extern "C" void kernel_launch(void* const* d_in, const int* in_sizes, int n_in,
                              void* d_out, int out_size, void* d_ws, size_t ws_size,
                              hipStream_t stream) {
  const float* x    = (const float*)d_in[0];   // (B,T,N,1)
  const float* ycov = (const float*)d_in[1];   // (B,T,N,1)
  const float* nemb = (const float*)d_in[2];   // (N,E)
  const float* egW  = (const float*)d_in[3];   // (195,128)
  const float* egb  = (const float*)d_in[4];   // (128,)
  const float* euW  = (const float*)d_in[5];   // (195,64)
  const float* eub  = (const float*)d_in[6];   // (64,)
  const float* dgW  = (const float*)d_in[7];   // (198,128)
  const float* dgb  = (const float*)d_in[8];   // (128,)
  const float* duW  = (const float*)d_in[9];   // (198,64)
  const float* dub  = (const float*)d_in[10];  // (64,)
  const float* pW   = (const float*)d_in[11];  // (64,1)
  const float* pb   = (const float*)d_in[12];  // (1,)
  const float* hW   = (const float*)d_in[13];  // (64,10)
  const float* hb   = (const float*)d_in[14];  // (10,)
  float* out = (float*)d_out;                  // (B,T,N,1)

  // --- workspace carve-up (floats) ---
  float* W = (float*)d_ws;
  size_t off = 0;
  auto alloc = [&](size_t n) { float* p = W + off; off += n; return p; };
  const long long BNrows = (long long)BB * NN;        // 16384
  float* A1  = alloc((size_t)NN * NN);                //  1 MB static A
  float* A2  = alloc((size_t)NN * NN);                //  1 MB 2A^2 - I
  float* h   = alloc((size_t)BNrows * HH);            //  4 MB hidden state
  float* zr  = alloc((size_t)BNrows * 2 * HH);        //  8 MB gates
  float* hc  = alloc((size_t)BNrows * HH);            //  4 MB candidate
  float* xg  = alloc((size_t)BNrows * DEC_LD);        // ~12 MB (198 >= 195)
  float* Ed  = alloc((size_t)BNrows * EE);            //  dyn embeddings
  float* Ad  = alloc((size_t)BB * NN * NN);           // 32 MB dyn A
  float* Ad2 = alloc((size_t)BB * NN * NN);           // 32 MB dyn 2A^2-I
  float* go  = alloc((size_t)BNrows);                 //  decoder output feed
  (void)ws_size; (void)in_sizes; (void)n_in; (void)out_size;

  const long long NNb = (long long)NN * NN;

  // 1) Static adjacency + Chebyshev T2
  adj_softmax_kernel<<<dim3(NN, 1), 256, 0, stream>>>(nemb, A1, 0, 0);
  launch_gemm(stream, A1, NN, 0, A1, NN, 0, A2, NN, 0,
              NN, NN, NN, 1, OP_CHEB, nullptr);

  // 2) init h0 = 0, go0 = 0
  hipMemsetAsync(h, 0, (size_t)BNrows * HH * sizeof(float), stream);
  hipMemsetAsync(go, 0, (size_t)BNrows * sizeof(float), stream);

  const int ecat_blocks = (int)((BNrows * ENC_C + 255) / 256);
  const int dcat_blocks = (int)((BNrows * DEC_C + 255) / 256);
  const int comb_blocks = (int)((BNrows * HH + 255) / 256);

  // 3) Encoder: 12 GRU steps with static graphs
  for (int t = 0; t < TT; ++t) {
    // gate branch: inp = [x_t, h] into xg[..., 0:65] (identity term)
    enc_concat_kernel<<<ecat_blocks, 256, 0, stream>>>(x, h, zr, xg, t, 0);
    // xg[...,65:130] = A @ inp  ;  xg[...,130:195] = A2 @ inp   (batched over B)
    launch_gemm(stream, A1, NN, 0, xg, ENC_LD, (long long)NN * ENC_LD,
                xg + ENC_C, ENC_LD, (long long)NN * ENC_LD,
                NN, ENC_C, NN, BB, OP_NONE, nullptr);
    launch_gemm(stream, A2, NN, 0, xg, ENC_LD, (long long)NN * ENC_LD,
                xg + 2 * ENC_C, ENC_LD, (long long)NN * ENC_LD,
                NN, ENC_C, NN, BB, OP_NONE, nullptr);
    // zr = sigmoid(xg @ egW + egb)
    launch_gemm(stream, xg, ENC_LD, 0, egW, 2 * HH, 0, zr, 2 * HH, 0,
                (int)BNrows, 2 * HH, ENC_LD, 1, OP_SIG, egb);
    // candidate branch: cand = [x_t, z*h]
    enc_concat_kernel<<<ecat_blocks, 256, 0, stream>>>(x, h, zr, xg, t, 1);
    launch_gemm(stream, A1, NN, 0, xg, ENC_LD, (long long)NN * ENC_LD,
                xg + ENC_C, ENC_LD, (long long)NN * ENC_LD,
                NN, ENC_C, NN, BB, OP_NONE, nullptr);
    launch_gemm(stream, A2, NN, 0, xg, ENC_LD, (long long)NN * ENC_LD,
                xg + 2 * ENC_C, ENC_LD, (long long)NN * ENC_LD,
                NN, ENC_C, NN, BB, OP_NONE, nullptr);
    // hc = tanh(xg @ euW + eub)
    launch_gemm(stream, xg, ENC_LD, 0, euW, HH, 0, hc, HH, 0,
                (int)BNrows, HH, ENC_LD, 1, OP_TANH, eub);
    // h = r*h + (1-r)*hc
    gru_combine_kernel<<<comb_blocks, 256, 0, stream>>>(h, zr, hc);
  }

  // 4) Dynamic graphs from final hidden state
  edyn_kernel<<<(int)((BNrows * EE + 255) / 256), 256, 0, stream>>>(h, hW, hb, nemb, Ed);
  adj_softmax_kernel<<<dim3(NN, BB), 256, 0, stream>>>(Ed, Ad,
      (long long)NN * EE, NNb);
  launch_gemm(stream, Ad, NN, NNb, Ad, NN, NNb, Ad2, NN, NNb,
              NN, NN, NN, BB, OP_CHEB, nullptr);

  // 5) Decoder: 12 GRU steps with per-batch dynamic graphs
  for (int t = 0; t < TT; ++t) {
    dec_concat_kernel<<<dcat_blocks, 256, 0, stream>>>(go, ycov, h, zr, xg, t, 0);
    launch_gemm(stream, Ad, NN, NNb, xg, DEC_LD, (long long)NN * DEC_LD,
                xg + DEC_C, DEC_LD, (long long)NN * DEC_LD,
                NN, DEC_C, NN, BB, OP_NONE, nullptr);
    launch_gemm(stream, Ad2, NN, NNb, xg, DEC_LD, (long long)NN * DEC_LD,
                xg + 2 * DEC_C, DEC_LD, (long long)NN * DEC_LD,
                NN, DEC_C, NN, BB, OP_NONE, nullptr);
    launch_gemm(stream, xg, DEC_LD, 0, dgW, 2 * HH, 0, zr, 2 * HH, 0,
                (int)BNrows, 2 * HH, DEC_LD, 1, OP_SIG, dgb);
    dec_concat_kernel<<<dcat_blocks, 256, 0, stream>>>(go, ycov, h, zr, xg, t, 1);
    launch_gemm(stream, Ad, NN, NNb, xg, DEC_LD, (long long)NN * DEC_LD,
                xg + DEC_C, DEC_LD, (long long)NN * DEC_LD,
                NN, DEC_C, NN, BB, OP_NONE, nullptr);
    launch_gemm(stream, Ad2, NN, NNb, xg, DEC_LD, (long long)NN * DEC_LD,
                xg + 2 * DEC_C, DEC_LD, (long long)NN * DEC_LD,
                NN, DEC_C, NN, BB, OP_NONE, nullptr);
    launch_gemm(stream, xg, DEC_LD, 0, duW, HH, 0, hc, HH, 0,
                (int)BNrows, HH, DEC_LD, 1, OP_TANH, dub);
    gru_combine_kernel<<<comb_blocks, 256, 0, stream>>>(h, zr, hc);
    proj_kernel<<<(int)((BNrows + 255) / 256), 256, 0, stream>>>(h, pW, pb, go, out, t);
  }
}